// MultiRNNModel_64330020159577
// MI455X (gfx1250) — compile-verified
//
#include <hip/hip_runtime.h>
#include <hip/hip_bf16.h>

typedef __attribute__((ext_vector_type(2))) float v2f;
typedef __attribute__((ext_vector_type(8))) float v8f;

constexpr int kB   = 1048576;
constexpr int kN   = 5;
constexpr int kCLS = 24;
constexpr int kOUT = 5;

// staging tile strides (dwords)
constexpr int FS = 36;   // 144 B: 16B-aligned rows, odd-ish bank cycle of 16
constexpr int WS = 25;   // weight row stride, coprime with 64 banks

__device__ __forceinline__ float sigmoidf_(float x) {
    return 1.0f / (1.0f + __expf(-x));
}

__global__ __launch_bounds__(256)
void fused_rnn_mlp_kernel(const float* __restrict__ seqs,
                          const float* __restrict__ others,
                          const float* __restrict__ w_ih,
                          const float* __restrict__ b_ih,
                          const float* __restrict__ b_hh,
                          const float* __restrict__ w1,
                          const float* __restrict__ b1,
                          const float* __restrict__ w2,
                          const float* __restrict__ b2,
                          float* __restrict__ out)
{
    // ---- LDS: weights staged once per block; per-wave 32x36 staging tile ----
    __shared__ float s_w1[32 * WS];   // rows 24..31 zero (N padding for tile 1)
    __shared__ float s_b1[32];        // idx 24..31 zero
    __shared__ float s_w2[16 * WS];   // rows 5..15 zero (N padding)
    __shared__ float s_b2[16];        // idx 5..15 zero
    __shared__ float s_wih[80];       // w_ih[n][g][h]
    __shared__ float s_bg[40];        // b_ih + b_hh
    __shared__ __align__(16) float s_feat[8][32][FS];

    const int tid  = threadIdx.x;
    const int wave = tid >> 5;
    const int lane = tid & 31;

    for (int i = tid; i < 32 * WS; i += 256) {
        const int n = i / WS, k = i % WS;
        s_w1[i] = (n < kCLS && k < kCLS) ? w1[n * 24 + k] : 0.0f;
    }
    for (int i = tid; i < 16 * WS; i += 256) {
        const int n = i / WS, k = i % WS;
        s_w2[i] = (n < kOUT && k < kCLS) ? w2[n * 24 + k] : 0.0f;
    }
    if (tid < 32) s_b1[tid]  = (tid < kCLS) ? b1[tid] : 0.0f;
    if (tid < 16) s_b2[tid]  = (tid < kOUT) ? b2[tid] : 0.0f;
    if (tid < 80) s_wih[tid] = w_ih[tid];
    if (tid < 40) s_bg[tid]  = b_ih[tid] + b_hh[tid];
    __syncthreads();

    const int waveBase = blockIdx.x * 256 + wave * 32;
    float* sf = &s_feat[wave][0][0];   // [32][FS]

    // ---------- Phase 1: LSTM cell (h_prev = 0) + gather "others" ----------
    {
        const int b = waveBase + lane;
        float feat[24];
        #pragma unroll
        for (int n = 0; n < kN; ++n) {
            const float2 x = *(const float2*)(seqs + (size_t)n * kB * 2 + (size_t)b * 2);
            float g[8];
            #pragma unroll
            for (int q = 0; q < 8; ++q)
                g[q] = fmaf(x.x, s_wih[n * 16 + q * 2 + 0],
                       fmaf(x.y, s_wih[n * 16 + q * 2 + 1], s_bg[n * 8 + q]));
            #pragma unroll
            for (int h = 0; h < 2; ++h) {
                const float c = sigmoidf_(g[0 + h]) * tanhf(g[4 + h]);
                feat[n * 2 + h] = sigmoidf_(g[6 + h]) * tanhf(c);
            }
        }
        const float2* op2 = (const float2*)(others + (size_t)b * 14);
        #pragma unroll
        for (int q = 0; q < 7; ++q) {
            const float2 v = op2[q];
            feat[10 + 2 * q]     = v.x;
            feat[10 + 2 * q + 1] = v.y;
        }
        // 6 x ds_store_b128 (row base = lane*144 B, 16B aligned)
        float4* dst = (float4*)(sf + lane * FS);
        #pragma unroll
        for (int q = 0; q < 6; ++q)
            dst[q] = make_float4(feat[4 * q], feat[4 * q + 1],
                                 feat[4 * q + 2], feat[4 * q + 3]);
    }
    asm volatile("s_wait_dscnt 0x0" ::: "memory");

    // WMMA f32 16x16x4 fragment mapping (wave32):
    //  A (16x4):  lane m = lane&15 is the M row; VGPR j holds K = 2*(lane>>4)+j
    //  B (4x16):  lane n = lane&15 is the N col; VGPR j holds K = 2*(lane>>4)+j
    //  C/D:       VGPR v holds M = v + 8*(lane>>4), N = lane&15
    const int m     = lane & 15;
    const int halfK = lane >> 4;
    const int n0    = m;        // N-tile 0 column (hidden 0..15 / out 0..4)
    const int n1    = 16 + m;   // N-tile 1 column (hidden 16..23; 24..31 zero-padded)

    #pragma unroll
    for (int t = 0; t < 2; ++t) {
        const float* frow = sf + (t * 16 + m) * FS;

        // ------------- layer 1: hidden = relu(feats @ w1.T + b1) -------------
        v8f acc0 = {};
        v8f acc1 = {};
        #pragma unroll
        for (int c = 0; c < 6; ++c) {
            const int kk = 4 * c + 2 * halfK;
            v2f a   = { frow[kk],           frow[kk + 1] };
            v2f bt0 = { s_w1[n0 * WS + kk], s_w1[n0 * WS + kk + 1] };
            v2f bt1 = { s_w1[n1 * WS + kk], s_w1[n1 * WS + kk + 1] };
            acc0 = __builtin_amdgcn_wmma_f32_16x16x4_f32(false, a, false, bt0,
                                                         (short)0, acc0, false, false);
            acc1 = __builtin_amdgcn_wmma_f32_16x16x4_f32(false, a, false, bt1,
                                                         (short)0, acc1, false, false);
        }
        const float bb0 = s_b1[n0];
        const float bb1 = s_b1[n1];
        float h0[8], h1[8];
        #pragma unroll
        for (int v = 0; v < 8; ++v) {
            h0[v] = fmaxf(acc0[v] + bb0, 0.0f);
            h1[v] = fmaxf(acc1[v] + bb1, 0.0f);
        }
        asm volatile("" ::: "memory");   // don't hoist stores above fragment reads
        #pragma unroll
        for (int v = 0; v < 8; ++v) {
            const int row = t * 16 + v + 8 * halfK;
            sf[row * FS + n0] = h0[v];        // cols 0..15
            sf[row * FS + n1] = h1[v];        // cols 16..31 (24..31 = dead pad)
        }
        asm volatile("s_wait_dscnt 0x0" ::: "memory");

        // ------------- layer 2: out = sigmoid(hidden @ w2.T + b2) -------------
        v8f acc2 = {};
        #pragma unroll
        for (int c = 0; c < 6; ++c) {
            const int kk = 4 * c + 2 * halfK;
            v2f a  = { frow[kk],           frow[kk + 1] };   // hidden row (in-place)
            v2f bw = { s_w2[n0 * WS + kk], s_w2[n0 * WS + kk + 1] };
            acc2 = __builtin_amdgcn_wmma_f32_16x16x4_f32(false, a, false, bw,
                                                         (short)0, acc2, false, false);
        }
        const float bo = s_b2[n0];
        asm volatile("" ::: "memory");
        #pragma unroll
        for (int v = 0; v < 8; ++v) {
            const int row = t * 16 + v + 8 * halfK;
            // cols 0..4 are the real outputs; cols 5..15 are dead pad (never read)
            sf[row * FS + n0] = sigmoidf_(acc2[v] + bo);
        }
    }
    asm volatile("s_wait_dscnt 0x0" ::: "memory");

    // ---- coalesced store: this wave owns 32*5 = 160 contiguous output floats ----
    float* op = out + (size_t)waveBase * kOUT;
    #pragma unroll
    for (int q = 0; q < 5; ++q) {
        const int f = lane + 32 * q;
        op[f] = sf[(f / 5) * FS + (f % 5)];
    }
}

extern "C" void kernel_launch(void* const* d_in, const int* in_sizes, int n_in,
                              void* d_out, int out_size, void* d_ws, size_t ws_size,
                              hipStream_t stream) {
    (void)in_sizes; (void)n_in; (void)out_size; (void)d_ws; (void)ws_size;
    const float* seqs   = (const float*)d_in[0];
    const float* others = (const float*)d_in[1];
    const float* w_ih   = (const float*)d_in[2];
    // d_in[3] = w_hh is mathematically unused by the reference (h_prev == 0)
    const float* b_ih   = (const float*)d_in[4];
    const float* b_hh   = (const float*)d_in[5];
    const float* w1     = (const float*)d_in[6];
    const float* b1     = (const float*)d_in[7];
    const float* w2     = (const float*)d_in[8];
    const float* b2     = (const float*)d_in[9];
    float* out = (float*)d_out;

    dim3 grid(kB / 256);   // 4096 blocks x 256 threads; 32 samples per wave32
    fused_rnn_mlp_kernel<<<grid, 256, 0, stream>>>(
        seqs, others, w_ih, b_ih, b_hh, w1, b1, w2, b2, out);
}